// Model_64785286693222
// MI455X (gfx1250) — compile-verified
//
#include <hip/hip_runtime.h>
#include <hip/hip_bf16.h>
#include <math.h>

typedef __attribute__((ext_vector_type(2))) float v2f;
typedef __attribute__((ext_vector_type(8))) float v8f;
typedef int i32x4 __attribute__((vector_size(16)));

namespace {
constexpr int kB   = 4;
constexpr int kL   = 2048;
constexpr int kDin = 80;
constexpr int kD   = 128;
constexpr int kH   = 16;
constexpr int kDh  = 8;
constexpr int kWin = 64;
constexpr int kT   = kB * kL;          // 8192 tokens
constexpr int kE   = 3 * kD;           // 384 (qkv row)
}

// ---- async global->LDS copy (CDNA5), with safe fallback -------------------
#if __has_builtin(__builtin_amdgcn_global_load_async_to_lds_b128)
#define ASYNC_LDS 1
#else
#define ASYNC_LDS 0
#endif

__device__ __forceinline__ void cp16_g2l(const float* __restrict__ g,
                                         float* __restrict__ l) {
#if ASYNC_LDS
  __builtin_amdgcn_global_load_async_to_lds_b128(
      (__attribute__((address_space(1))) i32x4*)g,
      (__attribute__((address_space(3))) i32x4*)l, 0, 0);
#else
  *reinterpret_cast<float4*>(l) = *reinterpret_cast<const float4*>(g);
#endif
}

__device__ __forceinline__ void wait_async_copies() {
#if ASYNC_LDS
#if __has_builtin(__builtin_amdgcn_s_wait_asynccnt)
  __builtin_amdgcn_s_wait_asynccnt(0);
#else
  asm volatile("s_wait_asynccnt 0" ::: "memory");
#endif
#endif
}

// ---------------------------------------------------------------------------
// Block-cooperative staged GEMM: a block of MT*NT waves computes an
// (MT*16) x (NT*16) fp32 output tile of  C = A @ W^T  (W torch-layout [N][K]).
// A slice (MT*16 rows) and W slice (NT*16 rows) are staged into LDS (async
// b128 copies, ASYNCcnt), then each wave runs K/4 V_WMMA_F32_16X16X4_F32 ops
// with both fragments sourced from LDS (ds_load_b64).
//
// Fragment layout per WMMA (wave32):
//   A: lane&15 = m, (lane>>4)*2 = k-offset inside the 4-wide K step
//   B: lane&15 = n, same K striping (B[k][n] = W[n*K+k])
//   C/D: lane&15 = n, vgpr r -> m = r + 8*(lane>>4)
// ---------------------------------------------------------------------------
template <int KDIM, int MT, int NT>
__device__ __forceinline__ v8f staged_gemm(const float* __restrict__ A,
                                           const float* __restrict__ W,
                                           int tm0, int n00,
                                           float* sA, float* sW) {
  constexpr int LDP = KDIM + 4;                 // row pad: 16B-aligned rows,
  constexpr int KF4 = KDIM / 4;                 // 4-bank lane stride
  constexpr int NTHREADS = MT * NT * 32;
  const int tid = threadIdx.x;

  const float* gA = A + (size_t)tm0 * KDIM;
  const float* gW = W + (size_t)n00 * KDIM;
#pragma unroll
  for (int i = tid; i < MT * 16 * KF4; i += NTHREADS) {
    const int row = i / KF4, col = (i % KF4) * 4;
    cp16_g2l(gA + (size_t)row * KDIM + col, sA + row * LDP + col);
  }
#pragma unroll
  for (int i = tid; i < NT * 16 * KF4; i += NTHREADS) {
    const int row = i / KF4, col = (i % KF4) * 4;
    cp16_g2l(gW + (size_t)row * KDIM + col, sW + row * LDP + col);
  }
  wait_async_copies();
  __syncthreads();

  const int lane = tid & 31;
  const int wv   = tid >> 5;
  const int tmi  = wv / NT;
  const int tni  = wv % NT;
  const int r16  = lane & 15;
  const int koff = (lane >> 4) << 1;
  const float* ap = sA + (tmi * 16 + r16) * LDP + koff;
  const float* wp = sW + (tni * 16 + r16) * LDP + koff;

  v8f acc = {};
#pragma unroll
  for (int kb = 0; kb < KDIM; kb += 4) {
    const float2 av = *reinterpret_cast<const float2*>(ap + kb);
    const float2 bv = *reinterpret_cast<const float2*>(wp + kb);
    v2f a;  a[0] = av.x;  a[1] = av.y;
    v2f b;  b[0] = bv.x;  b[1] = bv.y;
    acc = __builtin_amdgcn_wmma_f32_16x16x4_f32(
        false, a, false, b, (short)0, acc, false, false);
  }
  return acc;
}

// ---- Stage 1: conv1d(k=1) + ReLU + BatchNorm(eval) + sinusoidal PE --------
__global__ __launch_bounds__(256) void k_conv_bn_pe(
    const float* __restrict__ x, const float* __restrict__ conv_w,
    const float* __restrict__ conv_b, const float* __restrict__ bn_g,
    const float* __restrict__ bn_b, const float* __restrict__ bn_rm,
    const float* __restrict__ bn_rv, float* __restrict__ out) {
  constexpr int MT = 2, NT = 4, LDP = kDin + 4;
  __shared__ __align__(16) float sA[MT * 16 * LDP];
  __shared__ __align__(16) float sW[NT * 16 * LDP];
  const int tm0 = blockIdx.x * (MT * 16);
  const int n00 = blockIdx.y * (NT * 16);
  v8f acc = staged_gemm<kDin, MT, NT>(x, conv_w, tm0, n00, sA, sW);

  const int lane = threadIdx.x & 31;
  const int wv   = threadIdx.x >> 5;
  const int d     = n00 + (wv % NT) * 16 + (lane & 15);
  const int mbase = tm0 + (wv / NT) * 16 + ((lane >> 4) << 3);

  const float cb = conv_b[d];
  const float sc = bn_g[d] * rsqrtf(bn_rv[d] + 1e-5f);
  const float sh = bn_b[d] - bn_rm[d] * sc;
  // pos_j = 10000^(j/128), j = d/2; even d -> sin, odd d -> cos
  const int   j        = d >> 1;
  const float inv_freq = 1.0f / powf(10000.0f, (float)j * (1.0f / 128.0f));
#pragma unroll
  for (int r = 0; r < 8; ++r) {
    const int t = mbase + r;
    const int l = t & (kL - 1);
    const float ang = (float)l * inv_freq;
    const float pe  = (d & 1) ? cosf(ang) : sinf(ang);
    float v = fmaxf(acc[r] + cb, 0.0f);
    out[(size_t)t * kD + d] = v * sc + sh + pe;
  }
}

// ---- Generic fused linear: out = [relu](A @ W^T + bias) -------------------
template <int KDIM, int NDIM, int MT, int NT, bool RELU>
__global__ __launch_bounds__(MT* NT * 32) void k_linear_s(
    const float* __restrict__ A, const float* __restrict__ W,
    const float* __restrict__ bias, float* __restrict__ out) {
  constexpr int LDP = KDIM + 4;
  __shared__ __align__(16) float sA[MT * 16 * LDP];
  __shared__ __align__(16) float sW[NT * 16 * LDP];
  const int tm0 = blockIdx.x * (MT * 16);
  const int n00 = blockIdx.y * (NT * 16);
  v8f acc = staged_gemm<KDIM, MT, NT>(A, W, tm0, n00, sA, sW);

  const int lane = threadIdx.x & 31;
  const int wv   = threadIdx.x >> 5;
  const int n     = n00 + (wv % NT) * 16 + (lane & 15);
  const int mbase = tm0 + (wv / NT) * 16 + ((lane >> 4) << 3);
  const float bv = bias[n];
#pragma unroll
  for (int r = 0; r < 8; ++r) {
    float v = acc[r] + bv;
    if (RELU) v = fmaxf(v, 0.0f);
    out[(size_t)(mbase + r) * NDIM + n] = v;
  }
}

// ---- Banded local attention, online softmax, diag excluded ----------------
// One thread per (b, h, q). qkv row = [q(128) | k(128) | v(128)].
__global__ __launch_bounds__(256) void k_attn(const float* __restrict__ qkv,
                                              float* __restrict__ ctx) {
  const int tid = blockIdx.x * blockDim.x + threadIdx.x;   // 2^17 threads
  const int l = tid & (kL - 1);
  const int h = (tid >> 11) & (kH - 1);
  const int b = tid >> 15;

  const float* base = qkv + (size_t)b * kL * kE + h * kDh;
  const float* qp   = base + (size_t)l * kE;
  const float4 qa = *reinterpret_cast<const float4*>(qp);
  const float4 qb = *reinterpret_cast<const float4*>(qp + 4);
  const float q[8] = {qa.x, qa.y, qa.z, qa.w, qb.x, qb.y, qb.z, qb.w};

  const int jlo = (l > kWin) ? (l - kWin) : 0;
  const int jhi = (l + kWin < kL - 1) ? (l + kWin) : (kL - 1);
  const float invs = 0.3535533905932738f;                  // 1/sqrt(8)

  float mmax = -3.0e38f, ssum = 0.0f;
  float acc[8] = {0.f, 0.f, 0.f, 0.f, 0.f, 0.f, 0.f, 0.f};
  for (int jj = jlo; jj <= jhi; ++jj) {
    if (jj == l) continue;                                 // diagonal masked
    const float* kp = base + (size_t)jj * kE + kD;
    const float4 ka = *reinterpret_cast<const float4*>(kp);
    const float4 kb = *reinterpret_cast<const float4*>(kp + 4);
    float s = q[0] * ka.x + q[1] * ka.y + q[2] * ka.z + q[3] * ka.w +
              q[4] * kb.x + q[5] * kb.y + q[6] * kb.z + q[7] * kb.w;
    s *= invs;
    const float mn   = fmaxf(mmax, s);
    const float corr = expf(mmax - mn);
    const float w    = expf(s - mn);
    mmax = mn;
    ssum = ssum * corr + w;
    const float* vp = kp + kD;
    const float4 va = *reinterpret_cast<const float4*>(vp);
    const float4 vb = *reinterpret_cast<const float4*>(vp + 4);
    acc[0] = acc[0] * corr + w * va.x;  acc[1] = acc[1] * corr + w * va.y;
    acc[2] = acc[2] * corr + w * va.z;  acc[3] = acc[3] * corr + w * va.w;
    acc[4] = acc[4] * corr + w * vb.x;  acc[5] = acc[5] * corr + w * vb.y;
    acc[6] = acc[6] * corr + w * vb.z;  acc[7] = acc[7] * corr + w * vb.w;
  }
  const float inv = 1.0f / ssum;
  float* cp = ctx + (size_t)(b * kL + l) * kD + h * kDh;
#pragma unroll
  for (int i = 0; i < 8; ++i) cp[i] = acc[i] * inv;
}

extern "C" void kernel_launch(void* const* d_in, const int* in_sizes, int n_in,
                              void* d_out, int out_size, void* d_ws, size_t ws_size,
                              hipStream_t stream) {
  (void)in_sizes; (void)n_in; (void)out_size; (void)ws_size;
  const float* x         = (const float*)d_in[0];
  const float* conv_w    = (const float*)d_in[1];
  const float* conv_b    = (const float*)d_in[2];
  const float* bn_g      = (const float*)d_in[3];
  const float* bn_b      = (const float*)d_in[4];
  const float* bn_rm     = (const float*)d_in[5];
  const float* bn_rv     = (const float*)d_in[6];
  const float* in_proj_w = (const float*)d_in[7];
  const float* in_proj_b = (const float*)d_in[8];
  const float* out_w     = (const float*)d_in[9];
  const float* out_b     = (const float*)d_in[10];
  const float* ff_w1     = (const float*)d_in[11];
  const float* ff_b1     = (const float*)d_in[12];
  const float* ff_w2     = (const float*)d_in[13];
  const float* ff_b2     = (const float*)d_in[14];
  const float* fc_w      = (const float*)d_in[15];
  const float* fc_b      = (const float*)d_in[16];
  float* out = (float*)d_out;

  // Workspace layout (floats): h[1M] | qkv[3M] | ctx[1M]; later stages reuse.
  float* h   = (float*)d_ws;
  float* qkv = h   + (size_t)kT * kD;       // 12 MB
  float* ctx = qkv + (size_t)kT * kE;       // 4 MB
  float* att = h;                           // reuse (h dead after qkv)
  float* f1  = qkv;                         // reuse (qkv dead after attn)
  float* f2  = ctx;                         // reuse (ctx dead after out-proj)

  constexpr int MROWS = kT / 32;            // 256 blocks along tokens (MT=2)
  const dim3 blk256(256);
  const dim3 blk320(320);                   // 10 waves for the 2x5 fc grid

  k_conv_bn_pe<<<dim3(MROWS, kD / 64), blk256, 0, stream>>>(
      x, conv_w, conv_b, bn_g, bn_b, bn_rm, bn_rv, h);
  k_linear_s<kD, kE, 2, 4, false><<<dim3(MROWS, kE / 64), blk256, 0, stream>>>(
      h, in_proj_w, in_proj_b, qkv);
  k_attn<<<dim3((kB * kH * kL) / 256), blk256, 0, stream>>>(qkv, ctx);
  k_linear_s<kD, kD, 2, 4, false><<<dim3(MROWS, kD / 64), blk256, 0, stream>>>(
      ctx, out_w, out_b, att);
  k_linear_s<kD, kD, 2, 4, true><<<dim3(MROWS, kD / 64), blk256, 0, stream>>>(
      att, ff_w1, ff_b1, f1);
  k_linear_s<kD, kD, 2, 4, false><<<dim3(MROWS, kD / 64), blk256, 0, stream>>>(
      f1, ff_w2, ff_b2, f2);
  k_linear_s<kD, kDin, 2, 5, false><<<dim3(MROWS, 1), blk320, 0, stream>>>(
      f2, fc_w, fc_b, out);
}